// EMA_12077448036948
// MI455X (gfx1250) — compile-verified
//
#include <hip/hip_runtime.h>
#include <stdint.h>
#include <stddef.h>

// Problem constants (match reference)
#define N_CLASSES 100000
#define DIM 512
#define BATCH 16384

typedef __attribute__((ext_vector_type(4))) float v4f;
typedef __attribute__((ext_vector_type(4))) int   v4i;

// Address-space-qualified pointer types for the gfx1250 async-to-LDS builtin.
// Param 1 is `int4 addrspace(1)*` (global), param 2 is an addrspace(3) (LDS)
// pointer — per hipcc's diagnostic from the previous round.
typedef __attribute__((address_space(1))) v4i* gbl_v4i_ptr;
typedef __attribute__((address_space(3))) v4i* lds_v4i_ptr;

// ---------------------------------------------------------------------------
// Phase A: initialize last-occurrence table in workspace (cannot rely on the
// harness poison value — it is not re-applied between timed replays).
// ---------------------------------------------------------------------------
__global__ void ema_init_lastpos(int* __restrict__ lastpos) {
    int t = blockIdx.x * blockDim.x + threadIdx.x;
    if (t < N_CLASSES) lastpos[t] = -1;
}

// ---------------------------------------------------------------------------
// Phase B: lastpos[cls] = max batch position where cls occurs (last-write-wins
// semantics of jax .at[].set / torch index_put).
// ---------------------------------------------------------------------------
__global__ void ema_scatter_last(const int* __restrict__ idx,
                                 int* __restrict__ lastpos) {
    int b = blockIdx.x * blockDim.x + threadIdx.x;
    if (b < BATCH) atomicMax(lastpos + idx[b], b);
}

// ---------------------------------------------------------------------------
// Phase C: out[b,:] = (a*centers[cls,:] + (1-a)*x[lastpos[cls],:]) * inv_c
// One 128-thread block per batch row; each thread moves one float4
// (128 * 16B = 2KB = one full row). cls/lastpos/counts loads are uniform
// per block -> scalar loads. Row data is staged through LDS with the CDNA5
// async global->LDS path (ASYNCcnt) when the builtin is available.
// ---------------------------------------------------------------------------
__global__ __launch_bounds__(128) void ema_gather_out(
    const int*   __restrict__ idx,
    const float* __restrict__ x,
    const float* __restrict__ centers,
    const float* __restrict__ counts,
    const int*   __restrict__ lastpos,
    float*       __restrict__ out)
{
    const float kAlpha         = 0.95f;
    const float kOneMinusAlpha = 0.05f;
    const float kLogAlpha      = -0.05129329438755058f; // ln(0.95)

    const int b   = blockIdx.x;
    const int cls = idx[b];            // uniform -> s_load
    const int lb  = lastpos[cls];      // uniform
    const float cnt   = counts[cls] + 1.0f;
    const float inv_c = 1.0f / (1.0f - __expf(kLogAlpha * cnt));

    const int d = threadIdx.x << 2;    // 4 floats per thread
    const float* cptr = centers + (size_t)cls * DIM + d;
    const float* xptr = x       + (size_t)lb  * DIM + d;

    v4f cv, xv;
#if defined(__gfx1250__) && \
    __has_builtin(__builtin_amdgcn_global_load_async_to_lds_b128) && \
    __has_builtin(__builtin_amdgcn_s_wait_asynccnt)
    // CDNA5 async path: per-lane 16B global->LDS DMA, tracked by ASYNCcnt.
    // Each thread reads back only the slot it requested, so a single
    // s_wait_asynccnt 0 suffices (no workgroup barrier needed).
    __shared__ v4f ldsC[128];
    __shared__ v4f ldsX[128];
    __builtin_amdgcn_global_load_async_to_lds_b128(
        (gbl_v4i_ptr)(void*)cptr,
        (lds_v4i_ptr)(void*)&ldsC[threadIdx.x],
        0, 0);
    __builtin_amdgcn_global_load_async_to_lds_b128(
        (gbl_v4i_ptr)(void*)xptr,
        (lds_v4i_ptr)(void*)&ldsX[threadIdx.x],
        0, 0);
    __builtin_amdgcn_s_wait_asynccnt(0);
    cv = ldsC[threadIdx.x];
    xv = ldsX[threadIdx.x];
#else
    cv = *(const v4f*)cptr;
    xv = *(const v4f*)xptr;
#endif

    v4f o = (cv * kAlpha + xv * kOneMinusAlpha) * inv_c;
    // Output is streamed, never re-read: nontemporal store (TH=NT) keeps it
    // from polluting WGP$/L2.
    __builtin_nontemporal_store(o, (v4f*)(out + (size_t)b * DIM + d));
}

// ---------------------------------------------------------------------------
extern "C" void kernel_launch(void* const* d_in, const int* in_sizes, int n_in,
                              void* d_out, int out_size, void* d_ws, size_t ws_size,
                              hipStream_t stream)
{
    const int*   idx     = (const int*)  d_in[0];  // i       [BATCH]
    const float* x       = (const float*)d_in[1];  // x       [BATCH, DIM]
    const float* centers = (const float*)d_in[2];  // centers [N_CLASSES, DIM]
    const float* counts  = (const float*)d_in[3];  // counts  [N_CLASSES]
    float*       out     = (float*)d_out;          // [BATCH, DIM]
    int*         lastpos = (int*)d_ws;             // N_CLASSES ints (400 KB)

    ema_init_lastpos<<<(N_CLASSES + 255) / 256, 256, 0, stream>>>(lastpos);
    ema_scatter_last<<<(BATCH + 255) / 256, 256, 0, stream>>>(idx, lastpos);
    ema_gather_out<<<BATCH, 128, 0, stream>>>(idx, x, centers, counts, lastpos, out);
}